// PredictiveStateLexicon_23338852287190
// MI455X (gfx1250) — compile-verified
//
#include <hip/hip_runtime.h>
#include <hip/hip_bf16.h>
#include <cstddef>

// ---------------------------------------------------------------------------
// PredictiveStateLexicon for MI455X (gfx1250, wave32, WMMA)
// Sizes: B=8, T=1024, D=1024, S=256, RH=256, NS=16, M=B*T=8192
// ---------------------------------------------------------------------------

typedef float v2f __attribute__((ext_vector_type(2)));
typedef float v8f __attribute__((ext_vector_type(8)));

#define M_TOK   8192
#define D_MODEL 1024
#define S_DIM   256
#define T_LEN   1024

static __device__ __forceinline__ v8f wmma_f32(v2f a, v2f b, v8f c) {
  // V_WMMA_F32_16X16X4_F32: D = A(16x4) * B(4x16) + C(16x16), fp32 throughout.
  return __builtin_amdgcn_wmma_f32_16x16x4_f32(false, a, false, b, (short)0, c,
                                               false, false);
}

// ---------------------------------------------------------------------------
// K1: D[M,N] = A[M,K] @ W[K,N], fp32 WMMA.
// Block = 256 threads (8 waves). Block tile 128(M) x 64(N); each wave owns a
// 16x64 strip (4 accumulators), reusing its A fragment across 4 column tiles.
// W staged through LDS in K-chunks of 32 in FRAGMENT-READY pair layout:
//   Bs2[(k>>1)*64 + n] = { W[k][n], W[k+1][n] }
// so each lane's B fragment is a single aligned ds_load_b64 (no repack movs).
// Fragment layouts per CDNA5 ISA 7.12.2:
//   A 16x4 : lane(l&15)=row, VGPR v holds K = k0 + v + 2*(l>>4)  (b64 load)
//   B  4x16: lane(l&15)=col, VGPR v holds K = k0 + v + 2*(l>>4)
//   C/D    : lane(l&15)=col, VGPR v holds row v + 8*(l>>4)
// ---------------------------------------------------------------------------
__global__ __launch_bounds__(256) void wmma_gemm_f32_kernel(
    const float* __restrict__ A, const float* __restrict__ W,
    float* __restrict__ D, int M, int N, int K) {
  __shared__ v2f Bs2[16 * 64];             // 16 k-pairs x 64 cols = 8 KB
  const int tid   = threadIdx.x;
  const int lane  = tid & 31;
  const int wv    = tid >> 5;              // 0..7
  const int rowb  = blockIdx.y * 128 + wv * 16;
  const int colb  = blockIdx.x * 64;
  const int mloc  = lane & 15;
  const int khalf = lane >> 4;             // 0 or 1

  const v8f vzero = {0.f, 0.f, 0.f, 0.f, 0.f, 0.f, 0.f, 0.f};
  v8f acc[4];
  for (int t = 0; t < 4; ++t) acc[t] = vzero;

  const float* Arow = A + (size_t)(rowb + mloc) * K;

  for (int k0 = 0; k0 < K; k0 += 32) {
    __syncthreads();
    // cooperative stage of W[k0..k0+31][colb..colb+63] into pair layout
    for (int i = tid; i < 32 * 64; i += 256) {
      const int kk = i >> 6, c = i & 63;
      Bs2[(kk >> 1) * 64 + c][kk & 1] = W[(size_t)(k0 + kk) * N + colb + c];
    }
    __syncthreads();
#pragma unroll
    for (int kk = 0; kk < 8; ++kk) {
      const int k = k0 + kk * 4;
      const v2f a = *(const v2f*)(Arow + k + 2 * khalf);   // global_load_b64
      const int pair = kk * 2 + khalf;                     // (kk*4+2*khalf)>>1
#pragma unroll
      for (int t = 0; t < 4; ++t) {
        const v2f b = Bs2[pair * 64 + t * 16 + mloc];      // ds_load_b64
        acc[t] = wmma_f32(a, b, acc[t]);
      }
    }
  }

  for (int t = 0; t < 4; ++t)
    for (int v = 0; v < 8; ++v)
      D[(size_t)(rowb + v + 8 * khalf) * N + colb + t * 16 + mloc] = acc[t][v];
}

// ---------------------------------------------------------------------------
// Pack a 256x256 weight into fragment-ready pairs:
//   pW[(k>>1)*256 + n] = { W[k][n], W[k+1][n] }
// One-time cost; makes the sequential recurrence's B loads single b64 ops.
// ---------------------------------------------------------------------------
__global__ void pack_pairs_kernel(const float* __restrict__ W,
                                  v2f* __restrict__ pW) {
  const int idx = blockIdx.x * blockDim.x + threadIdx.x;  // 0 .. 128*256-1
  if (idx >= 128 * 256) return;
  const int p = idx >> 8;        // k pair
  const int n = idx & 255;
  v2f v;
  v[0] = W[(size_t)(2 * p) * 256 + n];
  v[1] = W[(size_t)(2 * p + 1) * 256 + n];
  pW[idx] = v;
}

// ---------------------------------------------------------------------------
// K2: GRU recurrence. One workgroup, 16 waves; all 8 batches share the M=16
// WMMA tile (rows 8..15 stay zero). Waves 0..7: H@Wsh for cols [32w,32w+32);
// waves 8..15: H@Wgh for the same columns, published via LDS. XP loads for
// step t are hoisted above the WMMA chain to overlap their latency.
// prefix[b,t,:] = H before the update.
// ---------------------------------------------------------------------------
__global__ __launch_bounds__(512) void recurrence_kernel(
    const float* __restrict__ XPs, const float* __restrict__ XPg,
    const v2f* __restrict__ pWsh, const v2f* __restrict__ pWgh,
    float* __restrict__ prefix) {
  __shared__ float Hs[16][S_DIM];   // hidden state (rows 0..7 live, 8..15 zero)
  __shared__ float Gb[8][S_DIM];    // gate pre-activations from waves 8..15

  const int tid    = threadIdx.x;
  const int lane   = tid & 31;
  const int wv     = tid >> 5;      // 0..15
  const bool isG   = wv >= 8;
  const int colb   = (wv & 7) * 32;
  const int mloc   = lane & 15;
  const int khalf  = lane >> 4;
  const v2f* Wm    = isG ? pWgh : pWsh;
  const bool updThread = (!isG) && (khalf == 0);

  for (int i = tid; i < 16 * S_DIM; i += 512) ((float*)Hs)[i] = 0.f;
  __syncthreads();

  const v8f vzero = {0.f, 0.f, 0.f, 0.f, 0.f, 0.f, 0.f, 0.f};

  for (int t = 0; t < T_LEN; ++t) {
    // Hoist the per-step input-projection loads (independent of H) so their
    // global latency overlaps the WMMA chain below.
    float xs[8][2], xg[8][2];
    if (updThread) {
#pragma unroll
      for (int v = 0; v < 8; ++v) {
        const size_t idx0 = ((size_t)v * T_LEN + t) * S_DIM + colb + mloc;
        xs[v][0] = XPs[idx0];
        xs[v][1] = XPs[idx0 + 16];
        xg[v][0] = XPg[idx0];
        xg[v][1] = XPg[idx0 + 16];
      }
    }

    v8f acc0 = vzero, acc1 = vzero;
#pragma unroll 4
    for (int k = 0; k < S_DIM; k += 4) {
      const int kr = k + 2 * khalf;                       // even
      const v2f a  = *(const v2f*)&Hs[mloc][kr];          // ds_load_b64
      const int pr = (k >> 1) + khalf;
      const v2f b0 = Wm[pr * 256 + colb + mloc];          // global_load_b64
      const v2f b1 = Wm[pr * 256 + colb + 16 + mloc];
      acc0 = wmma_f32(a, b0, acc0);
      acc1 = wmma_f32(a, b1, acc1);
    }
    if (isG && khalf == 0) {  // rows 0..7 live in lanes 0..15, VGPR v = batch
#pragma unroll
      for (int v = 0; v < 8; ++v) {
        Gb[v][colb + mloc]      = acc0[v];
        Gb[v][colb + 16 + mloc] = acc1[v];
      }
    }
    __syncthreads();
    if (updThread) {
#pragma unroll
      for (int v = 0; v < 8; ++v) {        // v == batch index
#pragma unroll
        for (int half = 0; half < 2; ++half) {
          const int n = colb + half * 16 + mloc;
          const size_t idx = ((size_t)v * T_LEN + t) * S_DIM + n;
          const float s_pre = (half ? acc1[v] : acc0[v]) + xs[v][half];
          const float g_pre = Gb[v][n] + xg[v][half];
          const float hold  = Hs[v][n];
          const float prop  = tanhf(s_pre);
          const float gate  = 1.f / (1.f + expf(-g_pre));
          prefix[idx] = hold;               // state BEFORE consuming token t
          Hs[v][n] = gate * prop + (1.f - gate) * hold;
        }
      }
    }
    __syncthreads();
  }
}

// ---------------------------------------------------------------------------
// K3: compact active tokens (slot >= 0). Expected ~0.13% of 8192 tokens.
// ---------------------------------------------------------------------------
__global__ void compact_kernel(const int* __restrict__ ids,
                               const int* __restrict__ t2s,
                               int* __restrict__ list_tok,
                               int* __restrict__ list_slot,
                               int* __restrict__ count) {
  const int tok = blockIdx.x * blockDim.x + threadIdx.x;
  if (tok >= M_TOK) return;
  const int slot = t2s[ids[tok]];
  if (slot >= 0) {
    const int idx = atomicAdd(count, 1);
    list_tok[idx]  = tok;
    list_slot[idx] = slot;
  }
}

// ---------------------------------------------------------------------------
// K4: bulk out = base (the 64 MB memory floor), float4 moves.
// ---------------------------------------------------------------------------
__global__ void copy_base_kernel(const float4* __restrict__ base,
                                 float4* __restrict__ out, int n4) {
  const int i = blockIdx.x * blockDim.x + threadIdx.x;
  if (i < n4) out[i] = base[i];
}

// ---------------------------------------------------------------------------
// K5: router + delta mix, active tokens only. Fixed grid (capacity M_TOK);
// blocks beyond *count retire immediately — graph-capture safe, deterministic.
// ---------------------------------------------------------------------------
__global__ __launch_bounds__(256) void router_mix_kernel(
    const float* __restrict__ prefix, const float* __restrict__ base,
    const float* __restrict__ Wrh, const float* __restrict__ Wro,
    const float* __restrict__ delta, const int* __restrict__ list_tok,
    const int* __restrict__ list_slot, const int* __restrict__ count,
    float* __restrict__ out, float* __restrict__ entsum) {
  if ((int)blockIdx.x >= *count) return;
  const int tok  = list_tok[blockIdx.x];
  const int slot = list_slot[blockIdx.x];
  const int i    = threadIdx.x;

  __shared__ float hv[256];
  __shared__ float lg[16];
  __shared__ float pr[16];

  const float* pj = prefix + (size_t)tok * S_DIM;
  const float* xj = base + (size_t)tok * D_MODEL;
  float acc = 0.f;
  for (int j = 0; j < S_DIM; ++j)   acc += pj[j] * Wrh[(size_t)j * 256 + i];
  for (int j = 0; j < D_MODEL; ++j) acc += xj[j] * Wrh[(size_t)(S_DIM + j) * 256 + i];
  hv[i] = tanhf(acc);
  __syncthreads();

  if (i < 16) {
    float a = 0.f;
    for (int j = 0; j < 256; ++j) a += hv[j] * Wro[j * 16 + i];
    lg[i] = a;
  }
  __syncthreads();

  if (i == 0) {
    float mx = lg[0];
    for (int n = 1; n < 16; ++n) mx = fmaxf(mx, lg[n]);
    float s = 0.f, e[16];
    for (int n = 0; n < 16; ++n) { e[n] = expf(lg[n] - mx); s += e[n]; }
    float ent = 0.f;
    const float inv = 1.f / s;
    for (int n = 0; n < 16; ++n) {
      const float p = e[n] * inv;
      pr[n] = p;
      ent -= p * logf(fmaxf(p, 1e-8f));
    }
    atomicAdd(entsum, ent);
  }
  __syncthreads();

  const float* dl = delta + (size_t)slot * 16 * D_MODEL;
#pragma unroll
  for (int r = 0; r < 4; ++r) {
    const int d = i + r * 256;
    float m = 0.f;
    for (int n = 0; n < 16; ++n) m += pr[n] * dl[n * D_MODEL + d];
    out[(size_t)tok * D_MODEL + d] = base[(size_t)tok * D_MODEL + d] + 0.25f * m;
  }
}

__global__ void init_accum_kernel(int* count, float* entsum) {
  if (threadIdx.x == 0) { *count = 0; *entsum = 0.f; }
}

__global__ void finalize_kernel(const int* __restrict__ count,
                                const float* __restrict__ entsum,
                                float* __restrict__ out) {
  const int c = *count;
  out[8388608] = (c > 0) ? (*entsum) / (float)c : 0.f;
  out[8388609] = (float)c / 8192.f;
}

// ---------------------------------------------------------------------------
extern "C" void kernel_launch(void* const* d_in, const int* in_sizes, int n_in,
                              void* d_out, int out_size, void* d_ws,
                              size_t ws_size, hipStream_t stream) {
  (void)in_sizes; (void)n_in; (void)out_size; (void)ws_size;
  const int*   ids   = (const int*)d_in[0];
  const float* X     = (const float*)d_in[1];   // [8,1024,1024]
  const int*   t2s   = (const int*)d_in[2];
  const float* Wsi   = (const float*)d_in[3];   // [1024,256]
  const float* Wsh   = (const float*)d_in[4];   // [256,256]
  const float* Wgi   = (const float*)d_in[5];   // [1024,256]
  const float* Wgh   = (const float*)d_in[6];   // [256,256]
  const float* Wrh   = (const float*)d_in[7];   // [1280,256]
  const float* Wro   = (const float*)d_in[8];   // [256,16]
  const float* delta = (const float*)d_in[9];   // [64,16,1024]
  float* out = (float*)d_out;

  float* ws      = (float*)d_ws;
  float* XPs     = ws;                           // [8192,256]
  float* XPg     = ws + 2097152;                 // [8192,256]
  float* prefix  = ws + 4194304;                 // [8192,256]
  v2f*   pWsh    = (v2f*)(ws + 6291456);         // [128*256] pairs (64K floats)
  v2f*   pWgh    = (v2f*)(ws + 6357248);         // wait-free separate region
  int*   ltok    = (int*)(ws + 6422528);         // [8192]
  int*   lslot   = ltok + M_TOK;                 // [8192]
  int*   count   = lslot + M_TOK;
  float* entsum  = (float*)(count + 1);

  init_accum_kernel<<<1, 32, 0, stream>>>(count, entsum);

  // One-time fragment-ready packing of the recurrence weights
  pack_pairs_kernel<<<128, 256, 0, stream>>>(Wsh, pWsh);
  pack_pairs_kernel<<<128, 256, 0, stream>>>(Wgh, pWgh);

  // Input projections: fp32 WMMA GEMMs [8192,1024] x [1024,256]
  dim3 gblk(256);
  dim3 ggrid(256 / 64, M_TOK / 128);
  wmma_gemm_f32_kernel<<<ggrid, gblk, 0, stream>>>(X, Wsi, XPs,
                                                   M_TOK, S_DIM, D_MODEL);
  wmma_gemm_f32_kernel<<<ggrid, gblk, 0, stream>>>(X, Wgi, XPg,
                                                   M_TOK, S_DIM, D_MODEL);

  // Sequential GRU recurrence (single WGP, H in LDS, WMMA per step)
  recurrence_kernel<<<1, 512, 0, stream>>>(XPs, XPg, pWsh, pWgh, prefix);

  // Active-token compaction (independent of recurrence)
  compact_kernel<<<M_TOK / 256, 256, 0, stream>>>(ids, t2s, ltok, lslot, count);

  // Bulk output = base
  copy_base_kernel<<<M_TOK, 256, 0, stream>>>((const float4*)X, (float4*)out,
                                              M_TOK * D_MODEL / 4);

  // Router + delta mix for the ~0.13% active tokens only
  router_mix_kernel<<<M_TOK, 256, 0, stream>>>(prefix, X, Wrh, Wro, delta,
                                               ltok, lslot, count, out, entsum);

  finalize_kernel<<<1, 1, 0, stream>>>(count, entsum, out);
}